// StraightThroughNormal_10368051052806
// MI455X (gfx1250) — compile-verified
//
#include <hip/hip_runtime.h>
#include <hip/hip_bf16.h>
#include <stdint.h>

typedef __attribute__((ext_vector_type(2))) float v2f;
typedef __attribute__((ext_vector_type(8))) float v8f;

#define KN 32768   // columns (feature dim)
#define KB 1024    // batch rows

// ---------------------------------------------------------------------------
// Kernel A: fused streaming copy x->out + batch abs-column-sum via WMMA.
// One wave32 owns 16 columns. A = ones(16x4) so D = sum of the 4 B-rows,
// replicated across all 16 D rows; accumulate 1024 rows in 256 WMMAs.
// Block = 256 threads = 8 waves = 128 columns; grid = KN/128 blocks.
// Emits ac[col] = exp(-5*(0.97*activ + 0.03*mean|x|)) into ws.
// ---------------------------------------------------------------------------
__global__ void __launch_bounds__(256)
stn_reduce_copy_kernel(const float* __restrict__ x,
                       const float* __restrict__ activ,
                       float* __restrict__ out,
                       float* __restrict__ wsf) {
  const int tid  = threadIdx.x;
  const int lane = tid & 31;
  const int wave = tid >> 5;
  const int n    = lane & 15;   // column within the wave's 16-col tile
  const int half = lane >> 4;   // which K-pair this lane supplies
  const int col  = blockIdx.x * 128 + wave * 16 + n;

  v2f a; a.x = 1.0f; a.y = 1.0f;   // all-ones A: D[m,n] = sum_k B[k,n]
  v8f c = {};                       // f32 accumulator (16x16 D, rows identical)

  // Lane supplies B rows (b + 2*half) and (b + 2*half + 1) for its column.
  size_t idx = (size_t)(2 * half) * KN + (size_t)col;
  #pragma unroll 4
  for (int b = 0; b < KB; b += 4) {
    float x0 = x[idx];
    float x1 = x[idx + KN];
    out[idx]      = x0;           // streaming copy fused with the reduction
    out[idx + KN] = x1;
    v2f bv; bv.x = __builtin_fabsf(x0); bv.y = __builtin_fabsf(x1);
    // 8 args: (neg_a, A, neg_b, B, c_mod, C, reuse_a, reuse_b)
    c = __builtin_amdgcn_wmma_f32_16x16x4_f32(false, a, false, bv,
                                              (short)0, c, false, false);
    idx += (size_t)4 * KN;
  }

  // Every lane's c[0] = colsum for its column (all D rows identical; f32 C/D
  // layout: VGPR0 holds N = lane%16). Halves duplicate -> half 0 writes.
  if (half == 0) {
    float mean = c[0] * (1.0f / (float)KB);
    float an   = 0.97f * activ[col] + 0.03f * mean;
    wsf[16 + col] = __expf(-5.0f * an);
  }
}

// ---------------------------------------------------------------------------
// Kernel B: single block, 1024 threads. Computes s = sum(ac), replaces
// ac[0] with 4000*s, builds the inclusive CDF in place (deterministic:
// LDS Hillis-Steele scan, no atomics). wsf[0] <- grand total.
// ---------------------------------------------------------------------------
__global__ void __launch_bounds__(1024)
stn_cdf_kernel(float* __restrict__ wsf) {
  const int PER = KN / 1024;             // 32 elements per thread
  __shared__ float s[1024];
  __shared__ float s_ac0;
  float* ac = wsf + 16;

  const int tid  = threadIdx.x;
  const int base = tid * PER;
  float local[32];
  float lsum = 0.0f;
  #pragma unroll
  for (int k = 0; k < PER; ++k) { local[k] = ac[base + k]; lsum += local[k]; }
  s[tid] = lsum;
  if (tid == 0) s_ac0 = local[0];        // stash original ac[0] before overwrite
  __syncthreads();

  // inclusive scan of per-thread sums
  for (int off = 1; off < 1024; off <<= 1) {
    float v   = s[tid];
    float add = (tid >= off) ? s[tid - off] : 0.0f;
    __syncthreads();
    s[tid] = v + add;
    __syncthreads();
  }

  const float total0 = s[1023];              // s = sum(ac) before replacement
  const float v0new  = 4000.0f * total0;     // ac[0] = s * 4000
  const float delta  = v0new - s_ac0;

  float run = (tid == 0) ? 0.0f : (s[tid - 1] + delta);
  #pragma unroll
  for (int k = 0; k < PER; ++k) {
    float v = (base + k == 0) ? v0new : local[k];
    run += v;
    ac[base + k] = run;                      // CDF overwrites ac in place
  }
  if (tid == 0) wsf[0] = total0 + delta;     // grand total of final weights
}

// ---------------------------------------------------------------------------
// Kernel C: 1024 categorical samples (counter-based splitmix64 hash ->
// uniform -> inverse-CDF binary search), scatter-add std into out[t, r]
// when r > 0. Rows are unique per thread -> plain RMW is race-free.
// ---------------------------------------------------------------------------
__global__ void stn_sample_kernel(const float* __restrict__ wsf,
                                  const float* __restrict__ stdp,
                                  float* __restrict__ out) {
  const int t = blockIdx.x * blockDim.x + threadIdx.x;
  if (t >= KB) return;

  const float* cdf  = wsf + 16;
  const float grand = wsf[0];

  uint64_t z = (uint64_t)(t + 1) * 0x9E3779B97F4A7C15ull + 42ull;
  z ^= z >> 30; z *= 0xBF58476D1CE4E5B9ull;
  z ^= z >> 27; z *= 0x94D049BB133111EBull;
  z ^= z >> 31;
  const float u = (float)(uint32_t)(z >> 40) * (1.0f / 16777216.0f); // [0,1)
  const float target = u * grand;

  int lo = 0, hi = KN - 1;                 // first j with cdf[j] > target
  while (lo < hi) {
    int mid = (lo + hi) >> 1;
    if (cdf[mid] > target) hi = mid; else lo = mid + 1;
  }
  if (lo > 0) {
    size_t o = (size_t)t * KN + (size_t)lo;
    out[o] += stdp[0];
  }
}

// ---------------------------------------------------------------------------
extern "C" void kernel_launch(void* const* d_in, const int* in_sizes, int n_in,
                              void* d_out, int out_size, void* d_ws, size_t ws_size,
                              hipStream_t stream) {
  (void)in_sizes; (void)n_in; (void)out_size; (void)ws_size;
  const float* x     = (const float*)d_in[0];   // [1024, 1, 32768] f32
  const float* activ = (const float*)d_in[1];   // [1, 32768] f32
  const float* stdp  = (const float*)d_in[2];   // [1] f32
  float* out = (float*)d_out;                   // [1024*32768] f32
  float* wsf = (float*)d_ws;                    // header(16) + ac/cdf(32768)

  stn_reduce_copy_kernel<<<KN / 128, 256, 0, stream>>>(x, activ, out, wsf);
  stn_cdf_kernel<<<1, 1024, 0, stream>>>(wsf);
  stn_sample_kernel<<<(KB + 255) / 256, 256, 0, stream>>>(wsf, stdp, out);
}